// GNNPolicy_8332236554386
// MI455X (gfx1250) — compile-verified
//
#include <hip/hip_runtime.h>
#include <math.h>

typedef __attribute__((ext_vector_type(2))) float v2f;
typedef __attribute__((ext_vector_type(8))) float v8f;

#define B_    16
#define NMAX_ 1024
#define FN_   128
#define EMAX_ 16384
#define FE_   8
#define NACT_ 64
#define KMSG  136   // FN + FE
#define KNODE 256   // FN + 128

// ---------------------------------------------------------------- zero scratch
__global__ void zero_ws_kernel(float* __restrict__ p, int n) {
  int i = blockIdx.x * blockDim.x + threadIdx.x;
  if (i < n) p[i] = 0.0f;
}

// ------------------------------------------------- message GEMM + scatter-add
// 64 threads = 2 waves. Each wave: 32 edges x 128 outputs, K=136 (34 x K4).
// M=32 blocking: each B fragment feeds TWO wmma ops (two A fragments), halving
// B-fragment global traffic per FLOP vs M=16.
__global__ __launch_bounds__(64)
void msg_kernel(const float* __restrict__ nf, const float* __restrict__ ef,
                const int* __restrict__ esrc, const int* __restrict__ edst,
                const int* __restrict__ esplit,
                const float* __restrict__ Wm, const float* __restrict__ bm,
                float* __restrict__ agg) {
  __shared__ float ldsA[64 * KMSG];   // staged [h_src | e] rows, 64 x 136
  __shared__ int   sDst[64];
  __shared__ float sMask[64];

  const int tid = threadIdx.x;
  const int edge0 = blockIdx.x * 64;          // global edge id base
  const int b  = edge0 / EMAX_;
  const int e0 = edge0 % EMAX_;

  // per-edge metadata (64 threads, 64 edges)
  const int eMine = e0 + tid;
  const int srcRow = b * NMAX_ + esrc[b * EMAX_ + eMine];
  sDst[tid]  = b * NMAX_ + edst[b * EMAX_ + eMine];
  sMask[tid] = (eMine < esplit[b]) ? 1.0f : 0.0f;

  // stage this thread's edge row A = [h[src], ef] into LDS
  for (int k = tid & 0; k < 0; ++k) {}  // (no-op; keep structure simple)
  {
    float* dst = &ldsA[tid * KMSG];
    const float* hsrc = nf + (size_t)srcRow * FN_;
    for (int k = 0; k < FN_; k += 4) {
      dst[k + 0] = hsrc[k + 0];
      dst[k + 1] = hsrc[k + 1];
      dst[k + 2] = hsrc[k + 2];
      dst[k + 3] = hsrc[k + 3];
    }
    const float* es = ef + ((size_t)b * EMAX_ + eMine) * FE_;
    for (int k = 0; k < FE_; ++k) dst[FN_ + k] = es[k];
  }
  __syncthreads();

  const int lane = tid & 31;
  const int wave = tid >> 5;
  const int ln   = lane & 15;
  const int hi   = lane >> 4;     // 0: K lo pair, 1: K hi pair
  const int koff = hi << 1;       // 0 or 2

  const float* aBase0 = &ldsA[(wave * 32 + ln) * KMSG];        // rows M=0..15
  const float* aBase1 = aBase0 + 16 * KMSG;                    // rows M=16..31

  v8f c0[8] = {};
  v8f c1[8] = {};
  for (int ks = 0; ks < KMSG / 4; ++ks) {
    const int kb = ks * 4;
    v2f a0 = *(const v2f*)(aBase0 + kb + koff);
    v2f a1 = *(const v2f*)(aBase1 + kb + koff);
    #pragma unroll
    for (int n = 0; n < 8; ++n) {
      const int col = n * 16 + ln;
      v2f bb;
      bb.x = Wm[(kb + koff    ) * 128 + col];
      bb.y = Wm[(kb + koff + 1) * 128 + col];
      c0[n] = __builtin_amdgcn_wmma_f32_16x16x4_f32(
          false, a0, false, bb, (short)0, c0[n], false, false);
      c1[n] = __builtin_amdgcn_wmma_f32_16x16x4_f32(
          false, a1, false, bb, (short)0, c1[n], false, false);
    }
  }

  // epilogue: bias + relu, mask padded edges, scatter-add into agg
  #pragma unroll
  for (int n = 0; n < 8; ++n) {
    const int col = n * 16 + ln;
    const float bias = bm[col];
    #pragma unroll
    for (int r = 0; r < 8; ++r) {
      const int m = r + 8 * hi;                 // row within 16-row subtile
      const int eA = wave * 32 + m;             // subtile 0: edges 0..15
      const int eB = eA + 16;                   // subtile 1: edges 16..31
      float vA = fmaxf(c0[n][r] + bias, 0.0f) * sMask[eA];
      float vB = fmaxf(c1[n][r] + bias, 0.0f) * sMask[eB];
      atomicAdd(&agg[(size_t)sDst[eA] * FN_ + col], vA);
      atomicAdd(&agg[(size_t)sDst[eB] * FN_ + col], vB);
    }
  }
}

// --------------------------- node update GEMM fused with per-graph sum-pooling
// 64 threads = 2 waves. Each wave: 32 nodes x 128 outputs, K=256 (64 x K4).
// hout is never materialized: masked rows are summed into embsum[b][128];
// both 16-row halves fold into a single atomic per column.
__global__ __launch_bounds__(64)
void node_kernel(const float* __restrict__ nf, const float* __restrict__ agg,
                 const int* __restrict__ nsplit,
                 const float* __restrict__ Wn, const float* __restrict__ bn,
                 float* __restrict__ embsum) {
  const int tid  = threadIdx.x;
  const int lane = tid & 31;
  const int wave = tid >> 5;
  const int ln   = lane & 15;
  const int hi   = lane >> 4;
  const int koff = hi << 1;

  const int row0  = blockIdx.x * 64 + wave * 32;   // first node row of tile
  const int b     = row0 / NMAX_;
  const int nloc0 = row0 % NMAX_;
  const int ns    = nsplit[b];
  const int rowA  = row0 + ln;                     // A rows for this lane
  const int rowB  = rowA + 16;

  v8f c0[8] = {};
  v8f c1[8] = {};
  for (int ks = 0; ks < KNODE / 4; ++ks) {
    const int kb = ks * 4;
    // A = [h | agg] row: contiguous halves, kb never straddles 128
    const float* base = (kb < FN_) ? nf : agg;
    const int    koffk = (kb < FN_) ? kb : (kb - FN_);
    v2f a0 = *(const v2f*)(base + (size_t)rowA * FN_ + koffk + koff);
    v2f a1 = *(const v2f*)(base + (size_t)rowB * FN_ + koffk + koff);
    #pragma unroll
    for (int n = 0; n < 8; ++n) {
      const int col = n * 16 + ln;
      v2f bb;
      bb.x = Wn[(kb + koff    ) * 128 + col];
      bb.y = Wn[(kb + koff + 1) * 128 + col];
      c0[n] = __builtin_amdgcn_wmma_f32_16x16x4_f32(
          false, a0, false, bb, (short)0, c0[n], false, false);
      c1[n] = __builtin_amdgcn_wmma_f32_16x16x4_f32(
          false, a1, false, bb, (short)0, c1[n], false, false);
    }
  }

  #pragma unroll
  for (int n = 0; n < 8; ++n) {
    const int col = n * 16 + ln;
    const float bias = bn[col];
    float s = 0.0f;
    #pragma unroll
    for (int r = 0; r < 8; ++r) {
      const int m = r + 8 * hi;                    // row within 16-row subtile
      const float nmA = ((nloc0 + m     ) < ns) ? 1.0f : 0.0f;
      const float nmB = ((nloc0 + m + 16) < ns) ? 1.0f : 0.0f;
      s += fmaxf(c0[n][r] + bias, 0.0f) * nmA;
      s += fmaxf(c1[n][r] + bias, 0.0f) * nmB;
    }
    atomicAdd(&embsum[b * FN_ + col], s);          // one atomic per col/lane
  }
}

// --------------------------------------------------- head: W_g, fc1, fc2, mask
// One block per batch row; tiny (~3 MFLOP total) -> plain VALU.
__global__ __launch_bounds__(256)
void head_kernel(const float* __restrict__ embsum, const int* __restrict__ nsplit,
                 const float* __restrict__ gf,
                 const float* __restrict__ Wg, const float* __restrict__ bg,
                 const float* __restrict__ W1, const float* __restrict__ b1,
                 const float* __restrict__ W2, const float* __restrict__ b2,
                 const float* __restrict__ am, float* __restrict__ out) {
  __shared__ float fe[256];
  __shared__ float h1[256];
  const int b = blockIdx.x;
  const int t = threadIdx.x;

  if (t < 128) {
    fe[t] = embsum[b * 128 + t] / (float)nsplit[b];       // mean-pooled nodes
  } else {
    const int n = t - 128;                                 // graph embedding
    float s = bg[n];
    for (int k = 0; k < 128; ++k) s += gf[b * 128 + k] * Wg[k * 128 + n];
    fe[t] = s;
  }
  __syncthreads();

  {
    float s = b1[t];
    for (int k = 0; k < 256; ++k) s += fe[k] * W1[k * 256 + t];
    h1[t] = fmaxf(s, 0.0f);
  }
  __syncthreads();

  if (t < NACT_) {
    float s = b2[t];
    for (int k = 0; k < 256; ++k) s += h1[k] * W2[k * NACT_ + t];
    const float a = am[b * NACT_ + t];
    const float lm = fmaxf(logf(a), -3.402823466e38f);     // clamp log(0)
    out[b * NACT_ + t] = s + lm;
  }
}

extern "C" void kernel_launch(void* const* d_in, const int* in_sizes, int n_in,
                              void* d_out, int out_size, void* d_ws, size_t ws_size,
                              hipStream_t stream) {
  const float* nf     = (const float*)d_in[0];
  const float* ef     = (const float*)d_in[1];
  const float* gf     = (const float*)d_in[2];
  const float* am     = (const float*)d_in[3];
  const int*   esrc   = (const int*)d_in[4];
  const int*   edst   = (const int*)d_in[5];
  const int*   nsplit = (const int*)d_in[6];
  const int*   esplit = (const int*)d_in[7];
  const float* Wm = (const float*)d_in[8];
  const float* bm = (const float*)d_in[9];
  const float* Wn = (const float*)d_in[10];
  const float* bn = (const float*)d_in[11];
  const float* Wg = (const float*)d_in[12];
  const float* bg = (const float*)d_in[13];
  const float* W1 = (const float*)d_in[14];
  const float* b1 = (const float*)d_in[15];
  const float* W2 = (const float*)d_in[16];
  const float* b2 = (const float*)d_in[17];
  float* out = (float*)d_out;

  float* agg    = (float*)d_ws;                         // [B*NMAX, 128] = 8 MB
  float* embsum = agg + (size_t)B_ * NMAX_ * FN_;       // [B, 128]

  const int zn = B_ * NMAX_ * FN_ + B_ * FN_;
  zero_ws_kernel<<<(zn + 255) / 256, 256, 0, stream>>>(agg, zn);

  msg_kernel<<<(B_ * EMAX_) / 64, 64, 0, stream>>>(
      nf, ef, esrc, edst, esplit, Wm, bm, agg);

  node_kernel<<<(B_ * NMAX_) / 64, 64, 0, stream>>>(
      nf, agg, nsplit, Wn, bn, embsum);

  head_kernel<<<B_, 256, 0, stream>>>(
      embsum, nsplit, gf, Wg, bg, W1, b1, W2, b2, am, out);
}